// DMTetGeometry_37065567764986
// MI455X (gfx1250) — compile-verified
//
#include <hip/hip_runtime.h>
#include <cmath>
#include <cstdint>
#include <cstddef>

#define TPB 256

#if defined(__HIP_DEVICE_COMPILE__) && __has_builtin(__builtin_amdgcn_global_load_async_to_lds_b32)
#define USE_ASYNC_LDS 1
#else
#define USE_ASYNC_LDS 0
#endif

#if defined(__HIP_DEVICE_COMPILE__) && __has_builtin(__builtin_amdgcn_global_load_async_to_lds_b128)
#define USE_ASYNC_LDS_B128 1
#else
#define USE_ASYNC_LDS_B128 0
#endif

typedef int vint4 __attribute__((vector_size(4 * sizeof(int))));

// ---- marching-tets tables ----
__device__ __constant__ int c_TT[16][6] = {
  {-1,-1,-1,-1,-1,-1},{1,0,2,-1,-1,-1},{4,0,3,-1,-1,-1},{1,4,2,1,3,4},
  {3,1,5,-1,-1,-1},{2,3,0,2,5,3},{1,4,0,1,5,4},{4,2,5,-1,-1,-1},
  {4,5,2,-1,-1,-1},{4,1,0,4,5,1},{3,2,0,3,5,2},{1,3,5,-1,-1,-1},
  {4,1,2,4,3,1},{3,0,4,-1,-1,-1},{2,0,1,-1,-1,-1},{-1,-1,-1,-1,-1,-1}};
__device__ __constant__ int c_NT[16] = {0,1,1,2,1,2,2,1,1,2,2,1,2,1,1,0};
__device__ __constant__ int c_E0[6] = {0,0,0,1,1,2};
__device__ __constant__ int c_E1[6] = {1,2,3,2,3,3};

// =====================================================================
// K1: per grid-vertex crossing mask over 7 canonical edge directions.
// Stages a contiguous sdf window into LDS via CDNA5 async-to-LDS DMA
// (B128 bulk + B32 tail), drained with s_wait_asynccnt.
// =====================================================================
__global__ void k_edge_mask(const float* __restrict__ sdf, int n, int S, int R,
                            uint32_t* __restrict__ cnt, uint8_t* __restrict__ mask) {
  extern __shared__ float lsdf[];
  const int S2 = S * S;
  const int win = TPB + S2 + S + 1;
  const int base = blockIdx.x * TPB;
  const int win4 = (win + 3) >> 2;   // 16B chunks
  for (int i4 = threadIdx.x; i4 < win4; i4 += TPB) {
    int i = i4 * 4;
    int g = base + i;
#if USE_ASYNC_LDS_B128
    if (g + 3 < n) {
      __builtin_amdgcn_global_load_async_to_lds_b128(
          (__attribute__((address_space(1))) vint4*)(sdf + g),
          (__attribute__((address_space(3))) vint4*)(lsdf + i), 0, 0);
      continue;
    }
#endif
    for (int k = 0; k < 4 && i + k < win; ++k) {
      int gg = g + k;
      if (gg >= n) gg = n - 1;
#if USE_ASYNC_LDS
      __builtin_amdgcn_global_load_async_to_lds_b32(
          (__attribute__((address_space(1))) int*)(sdf + gg),
          (__attribute__((address_space(3))) int*)(lsdf + i + k), 0, 0);
#else
      lsdf[i + k] = sdf[gg];
#endif
    }
  }
#if USE_ASYNC_LDS
#if __has_builtin(__builtin_amdgcn_s_wait_asynccnt)
  __builtin_amdgcn_s_wait_asynccnt(0);
#else
  asm volatile("s_wait_asynccnt 0" ::: "memory");
#endif
#endif
  __syncthreads();

  int vid = base + threadIdx.x;
  if (vid >= n) return;
  int z = vid % S;
  int y = (vid / S) % S;
  int x = vid / S2;
  bool o0 = lsdf[threadIdx.x] > 0.0f;
  int deltas[7] = {1, S, S + 1, S2, S2 + 1, S2 + S, S2 + S + 1};
  bool okx = (x < R), oky = (y < R), okz = (z < R);
  bool ok[7] = {okz, oky, oky && okz, okx, okx && okz, okx && oky, okx && oky && okz};
  unsigned m = 0;
#pragma unroll
  for (int i = 0; i < 7; ++i) {
    if (ok[i]) {
      bool o1 = lsdf[threadIdx.x + deltas[i]] > 0.0f;
      if (o1 != o0) m |= (1u << i);
    }
  }
  mask[vid] = (uint8_t)m;
  cnt[vid] = (uint32_t)__popc(m);
}

// =====================================================================
// Exclusive scan (hierarchical, 256 elems/block)
// =====================================================================
__global__ void k_scan_local(uint32_t* data, int n, uint32_t* sums) {
  __shared__ uint32_t sh[TPB];
  int i = blockIdx.x * TPB + threadIdx.x;
  uint32_t v = (i < n) ? data[i] : 0u;
  sh[threadIdx.x] = v;
  __syncthreads();
  for (int off = 1; off < TPB; off <<= 1) {
    uint32_t t = 0;
    if ((int)threadIdx.x >= off) t = sh[threadIdx.x - off];
    __syncthreads();
    if ((int)threadIdx.x >= off) sh[threadIdx.x] += t;
    __syncthreads();
  }
  uint32_t inc = sh[threadIdx.x];
  if (i < n) data[i] = inc - v;
  if (threadIdx.x == TPB - 1) sums[blockIdx.x] = inc;
}

__global__ void k_scan_addback(uint32_t* data, int n, const uint32_t* offs) {
  int i = blockIdx.x * TPB + threadIdx.x;
  if (i < n) data[i] += offs[blockIdx.x];
}

static void scanEx(uint32_t* d, int n, uint32_t* s1, uint32_t* s2, uint32_t* total,
                   hipStream_t st) {
  int b1 = (n + TPB - 1) / TPB;
  if (b1 == 1) {
    k_scan_local<<<1, TPB, 0, st>>>(d, n, total);
    return;
  }
  k_scan_local<<<b1, TPB, 0, st>>>(d, n, s1);
  int b2 = (b1 + TPB - 1) / TPB;
  if (b2 == 1) {
    k_scan_local<<<1, TPB, 0, st>>>(s1, b1, total);
  } else {
    k_scan_local<<<b2, TPB, 0, st>>>(s1, b1, s2);
    k_scan_local<<<1, TPB, 0, st>>>(s2, b2, total);
    k_scan_addback<<<b2, TPB, 0, st>>>(s1, b1, s2);
  }
  k_scan_addback<<<b1, TPB, 0, st>>>(d, n, s1);
}

// =====================================================================
// K3: emit interpolated crossing-edge vertices; record edge -> vert id.
// =====================================================================
__global__ void k_vert_emit(const float* __restrict__ sdf, const float* __restrict__ verts,
                            const float* __restrict__ deform, int n, int S, float dscale,
                            const uint32_t* __restrict__ vbase, const uint8_t* __restrict__ mask,
                            int* __restrict__ edgeVert, float* __restrict__ outv) {
  int vid = blockIdx.x * TPB + threadIdx.x;
  if (vid >= n) return;
  const int S2 = S * S;
  int deltas[7] = {1, S, S + 1, S2, S2 + 1, S2 + S, S2 + S + 1};
  unsigned m = mask[vid];
  uint32_t eid = vbase[vid];
  float s0 = 0.f, p0x = 0.f, p0y = 0.f, p0z = 0.f;
  if (m) {
    s0 = sdf[vid];
    p0x = verts[3 * vid + 0] + dscale * tanhf(deform[3 * vid + 0]);
    p0y = verts[3 * vid + 1] + dscale * tanhf(deform[3 * vid + 1]);
    p0z = verts[3 * vid + 2] + dscale * tanhf(deform[3 * vid + 2]);
  }
#pragma unroll
  for (int i = 0; i < 7; ++i) {
    int ev = -1;
    if (m & (1u << i)) {
      int v2 = vid + deltas[i];
      float s1 = sdf[v2];
      float denom = s0 - s1;            // s0 + (-s1), matches reference
      float w0 = -s1 / denom;
      float w1 = s0 / denom;
      float q0 = verts[3 * v2 + 0] + dscale * tanhf(deform[3 * v2 + 0]);
      float q1 = verts[3 * v2 + 1] + dscale * tanhf(deform[3 * v2 + 1]);
      float q2 = verts[3 * v2 + 2] + dscale * tanhf(deform[3 * v2 + 2]);
      outv[(size_t)3 * eid + 0] = p0x * w0 + q0 * w1;
      outv[(size_t)3 * eid + 1] = p0y * w0 + q1 * w1;
      outv[(size_t)3 * eid + 2] = p0z * w0 + q2 * w1;
      ev = (int)eid;
      eid++;
    }
    edgeVert[(size_t)vid * 7 + i] = ev;
  }
}

// =====================================================================
// K4: classify tets (tetindex, 1-tri / 2-tri flags for compaction)
// =====================================================================
__global__ void k_tet_class(const int* __restrict__ tets, const float* __restrict__ sdf, int T,
                            uint32_t* __restrict__ m1, uint32_t* __restrict__ m2,
                            uint8_t* __restrict__ tidx) {
  int t = blockIdx.x * TPB + threadIdx.x;
  if (t >= T) return;
  int4 v = ((const int4*)tets)[t];
  int ti = (sdf[v.x] > 0.f ? 1 : 0) | (sdf[v.y] > 0.f ? 2 : 0) |
           (sdf[v.z] > 0.f ? 4 : 0) | (sdf[v.w] > 0.f ? 8 : 0);
  tidx[t] = (uint8_t)ti;
  int nt = c_NT[ti];
  m1[t] = (nt == 1) ? 1u : 0u;
  m2[t] = (nt == 2) ? 1u : 0u;
}

// =====================================================================
// K6: emit faces (m1 tets first, then m2 tets x2) + uv_idx rows
// =====================================================================
__global__ void k_face_emit(const int* __restrict__ tets, const uint8_t* __restrict__ tidx,
                            const uint32_t* __restrict__ m1s, const uint32_t* __restrict__ m2s,
                            const int* __restrict__ edgeVert, const uint32_t* __restrict__ totals,
                            int T, int S, int NN, float* __restrict__ out) {
  int t = blockIdx.x * TPB + threadIdx.x;
  if (t >= T) return;
  int ti = tidx[t];
  int nt = c_NT[ti];
  if (nt == 0) return;
  uint32_t E = totals[0], C1 = totals[1], C2 = totals[2];
  size_t F = (size_t)C1 + 2 * (size_t)C2;
  size_t facesStart = (size_t)3 * E;
  size_t uvsStart = facesStart + 3 * F;
  size_t uvidxStart = uvsStart + (size_t)NN * NN * 8;

  int4 v = ((const int4*)tets)[t];
  int c[4] = {v.x, v.y, v.z, v.w};
  const int S2 = S * S;
  int ev[6];
#pragma unroll
  for (int e = 0; e < 6; ++e) {
    int a = c[c_E0[e]], b = c[c_E1[e]];
    if (a > b) { int tmp = a; a = b; b = tmp; }
    int d = b - a;
    int slot;
    if (d == 1) slot = 0;
    else if (d == S) slot = 1;
    else if (d == S + 1) slot = 2;
    else if (d == S2) slot = 3;
    else if (d == S2 + 1) slot = 4;
    else if (d == S2 + S) slot = 5;
    else slot = 6;
    ev[e] = edgeVert[(size_t)a * 7 + slot];
  }
  float q4 = (float)(4 * t);
  if (nt == 1) {
    size_t r = m1s[t];
    out[facesStart + 3 * r + 0] = (float)ev[c_TT[ti][0]];
    out[facesStart + 3 * r + 1] = (float)ev[c_TT[ti][1]];
    out[facesStart + 3 * r + 2] = (float)ev[c_TT[ti][2]];
    out[uvidxStart + 3 * r + 0] = q4;
    out[uvidxStart + 3 * r + 1] = q4 + 1.f;
    out[uvidxStart + 3 * r + 2] = q4 + 2.f;
  } else {
    size_t r = (size_t)C1 + 2 * (size_t)m2s[t];
    out[facesStart + 3 * r + 0] = (float)ev[c_TT[ti][0]];
    out[facesStart + 3 * r + 1] = (float)ev[c_TT[ti][1]];
    out[facesStart + 3 * r + 2] = (float)ev[c_TT[ti][2]];
    out[facesStart + 3 * (r + 1) + 0] = (float)ev[c_TT[ti][3]];
    out[facesStart + 3 * (r + 1) + 1] = (float)ev[c_TT[ti][4]];
    out[facesStart + 3 * (r + 1) + 2] = (float)ev[c_TT[ti][5]];
    out[uvidxStart + 3 * r + 0] = q4;          // tri_idx 0
    out[uvidxStart + 3 * r + 1] = q4 + 1.f;
    out[uvidxStart + 3 * r + 2] = q4 + 2.f;
    out[uvidxStart + 3 * (r + 1) + 0] = q4;    // tri_idx 1
    out[uvidxStart + 3 * (r + 1) + 1] = q4 + 2.f;
    out[uvidxStart + 3 * (r + 1) + 2] = q4 + 3.f;
  }
}

// =====================================================================
// K7: fill the static uv grid (N*N cells x 4 corners x 2 floats)
// =====================================================================
__global__ void k_uvs(const uint32_t* __restrict__ totals, int NN, float* __restrict__ out) {
  long long id = (long long)blockIdx.x * TPB + threadIdx.x;
  long long tot = (long long)NN * NN * 4;
  if (id >= tot) return;
  uint32_t E = totals[0], C1 = totals[1], C2 = totals[2];
  size_t F = (size_t)C1 + 2 * (size_t)C2;
  size_t uvsStart = (size_t)3 * E + 3 * F;
  long long cell = id >> 2;
  int r = (int)(id & 3);
  int i = (int)(cell / NN), j = (int)(cell % NN);
  float step = (float)((1.0 - 1.0 / (double)NN) / (double)(NN - 1));
  float xv = (float)j * step, yv = (float)i * step;
  float pad = (float)(0.9 / (double)NN);
  float u, w;
  switch (r) {
    case 0: u = xv;       w = yv;       break;
    case 1: u = xv + pad; w = yv;       break;
    case 2: u = xv + pad; w = yv + pad; break;
    default: u = xv;      w = yv + pad; break;
  }
  out[uvsStart + (size_t)id * 2 + 0] = u;
  out[uvsStart + (size_t)id * 2 + 1] = w;
}

// =====================================================================
extern "C" void kernel_launch(void* const* d_in, const int* in_sizes, int n_in,
                              void* d_out, int out_size, void* d_ws, size_t ws_size,
                              hipStream_t stream) {
  (void)n_in; (void)out_size; (void)ws_size;
  const float* verts  = (const float*)d_in[0];
  const float* sdf    = (const float*)d_in[1];
  const float* deform = (const float*)d_in[2];
  const int*   tets   = (const int*)d_in[3];
  int n = in_sizes[1];
  int T = in_sizes[3] / 4;
  int S = 1;
  while ((long long)S * S * S < (long long)n) ++S;   // S = R+1
  int R = S - 1;
  float dscale = 2.0f / (float)(R * 2);
  long long half = (2LL * T + 1) / 2;
  int NN = (int)std::ceil(std::sqrt((double)half));

  // workspace partition (256B aligned slices)
  uint8_t* w = (uint8_t*)d_ws;
  auto alloc = [&](size_t bytes) -> void* {
    void* p = (void*)w;
    w += (bytes + 255) & ~(size_t)255;
    return p;
  };
  uint32_t* vertScan = (uint32_t*)alloc(sizeof(uint32_t) * (size_t)n);
  uint8_t*  mask     = (uint8_t*)alloc((size_t)n);
  int*      edgeVert = (int*)alloc(sizeof(int) * (size_t)n * 7);
  uint32_t* m1s      = (uint32_t*)alloc(sizeof(uint32_t) * (size_t)T);
  uint32_t* m2s      = (uint32_t*)alloc(sizeof(uint32_t) * (size_t)T);
  uint8_t*  tidx     = (uint8_t*)alloc((size_t)T);
  int nb = (n + TPB - 1) / TPB, tb = (T + TPB - 1) / TPB;
  int maxb = nb > tb ? nb : tb;
  uint32_t* sums1  = (uint32_t*)alloc(sizeof(uint32_t) * (size_t)maxb);
  uint32_t* sums2  = (uint32_t*)alloc(sizeof(uint32_t) * TPB);
  uint32_t* totals = (uint32_t*)alloc(sizeof(uint32_t) * 4);

  int S2 = S * S;
  int win = TPB + S2 + S + 1;
  int winPad = (win + 3) & ~3;                       // 16B-aligned LDS window
  size_t shBytes = sizeof(float) * (size_t)winPad;

  k_edge_mask<<<nb, TPB, shBytes, stream>>>(sdf, n, S, R, vertScan, mask);
  scanEx(vertScan, n, sums1, sums2, totals + 0, stream);
  k_vert_emit<<<nb, TPB, 0, stream>>>(sdf, verts, deform, n, S, dscale, vertScan, mask,
                                      edgeVert, (float*)d_out);
  k_tet_class<<<tb, TPB, 0, stream>>>(tets, sdf, T, m1s, m2s, tidx);
  scanEx(m1s, T, sums1, sums2, totals + 1, stream);
  scanEx(m2s, T, sums1, sums2, totals + 2, stream);
  k_face_emit<<<tb, TPB, 0, stream>>>(tets, tidx, m1s, m2s, edgeVert, totals, T, S, NN,
                                      (float*)d_out);
  long long uvtot = (long long)NN * NN * 4;
  int ub = (int)((uvtot + TPB - 1) / TPB);
  k_uvs<<<ub, TPB, 0, stream>>>(totals, NN, (float*)d_out);
}